// GNN_30348238913564
// MI455X (gfx1250) — compile-verified
//
#include <hip/hip_runtime.h>
#include <hip/hip_bf16.h>

typedef float v2f __attribute__((ext_vector_type(2)));
typedef float v8f __attribute__((ext_vector_type(8)));

// ---------------------------------------------------------------------------
// Zero a float buffer
// ---------------------------------------------------------------------------
__global__ void sage_zero_kernel(float* __restrict__ p, long n) {
    long i = (long)blockIdx.x * blockDim.x + threadIdx.x;
    if (i < n) p[i] = 0.0f;
}

// ---------------------------------------------------------------------------
// Degree count: cnt[dst] += 1 per edge
// ---------------------------------------------------------------------------
__global__ void sage_count_kernel(const int* __restrict__ dst_idx,
                                  float* __restrict__ cnt, int E) {
    int e = blockIdx.x * blockDim.x + threadIdx.x;
    if (e < E) atomicAdd(&cnt[dst_idx[e]], 1.0f);
}

// ---------------------------------------------------------------------------
// Scatter-add features: agg[dst, :] += feat[src, :]
// One wave (32 lanes) per edge, each lane moves 4 consecutive floats (D=128).
// ---------------------------------------------------------------------------
__global__ void sage_scatter_kernel(const float* __restrict__ feat,
                                    const int* __restrict__ src_idx,
                                    const int* __restrict__ dst_idx,
                                    float* __restrict__ agg, int E) {
    int e = (int)(((long)blockIdx.x * blockDim.x + threadIdx.x) >> 5);
    if (e >= E) return;
    int lane = threadIdx.x & 31;
    int s = src_idx[e];
    int d = dst_idx[e];
    const float4 v = *(const float4*)(feat + (long)s * 128 + lane * 4);
    float* o = agg + (long)d * 128 + lane * 4;
    atomicAdd(o + 0, v.x);
    atomicAdd(o + 1, v.y);
    atomicAdd(o + 2, v.z);
    atomicAdd(o + 3, v.w);
}

// ---------------------------------------------------------------------------
// Fused SAGE dense stage:
//   out = (agg / max(cnt,1)) @ Wl  +  hin @ Wr  +  b    [optional ReLU]
// One wave per 16x16 output tile, fp32 WMMA (V_WMMA_F32_16X16X4_F32), K=128.
//
// f32 WMMA VGPR layout (ISA 7.12.2):
//   A 16x4 : lane L(0-15): {K=0,K=1} for row M=L; lane L+16: {K=2,K=3}
//   B 4x16 : lane L(0-15): rows {K=0,K=1} at col N=L; lane L+16: {K=2,K=3}
//   C 16x16: VGPR j, lanes 0-15 -> M=j, N=lane; lanes 16-31 -> M=j+8
// ---------------------------------------------------------------------------
__global__ void sage_gemm_wmma_kernel(const float* __restrict__ agg,
                                      const float* __restrict__ cnt,
                                      const float* __restrict__ hin,
                                      const float* __restrict__ Wl,
                                      const float* __restrict__ Wr,
                                      const float* __restrict__ bias,
                                      float* __restrict__ out,
                                      int N, int do_relu) {
    const int wave = (int)(((long)blockIdx.x * blockDim.x + threadIdx.x) >> 5);
    const int lane = threadIdx.x & 31;
    const int m0 = (wave >> 3) << 4;   // 8 column tiles of 16 across D_OUT=128
    const int n0 = (wave & 7) << 4;
    if (m0 >= N) return;

    const int mr    = lane & 15;   // A row within tile / B-C column within tile
    const int khalf = lane >> 4;   // 0: K=0,1   1: K=2,3

    const int  row  = m0 + mr;
    const float inv = 1.0f / fmaxf(cnt[row], 1.0f);

    const float* aRow = agg + (long)row * 128;
    const float* hRow = hin + (long)row * 128;

    v8f c = {};
    #pragma unroll 4
    for (int k = 0; k < 128; k += 4) {
        const int ka = k + 2 * khalf;                      // K index for this half-wave
        const float2 av = *(const float2*)(aRow + ka);     // A: {K=ka, K=ka+1}, row `row`
        const float2 hv = *(const float2*)(hRow + ka);
        v2f a, h, wl, wr;
        a.x = av.x * inv;  a.y = av.y * inv;
        h.x = hv.x;        h.y = hv.y;
        wl.x = Wl[(long)ka * 128 + n0 + mr];               // B: row ka, col n0+mr
        wl.y = Wl[(long)(ka + 1) * 128 + n0 + mr];
        wr.x = Wr[(long)ka * 128 + n0 + mr];
        wr.y = Wr[(long)(ka + 1) * 128 + n0 + mr];
        c = __builtin_amdgcn_wmma_f32_16x16x4_f32(false, a, false, wl,
                                                  (short)0, c, false, false);
        c = __builtin_amdgcn_wmma_f32_16x16x4_f32(false, h, false, wr,
                                                  (short)0, c, false, false);
    }

    const float bv = bias[n0 + mr];
    #pragma unroll
    for (int j = 0; j < 8; ++j) {
        const int om = m0 + j + 8 * khalf;
        float v = c[j] + bv;
        if (do_relu) v = fmaxf(v, 0.0f);
        out[(long)om * 128 + n0 + mr] = v;
    }
}

// ---------------------------------------------------------------------------
// Launch: zero -> count -> scatter(x) -> gemm1(relu) -> zero -> scatter(h) -> gemm2
// ---------------------------------------------------------------------------
extern "C" void kernel_launch(void* const* d_in, const int* in_sizes, int n_in,
                              void* d_out, int out_size, void* d_ws, size_t ws_size,
                              hipStream_t stream) {
    const float* x    = (const float*)d_in[0];
    const int*   ei   = (const int*)d_in[1];    // [2, E] flat: src then dst
    const float* Wl1  = (const float*)d_in[2];
    const float* Wr1  = (const float*)d_in[3];
    const float* b1   = (const float*)d_in[4];
    const float* Wl2  = (const float*)d_in[5];
    const float* Wr2  = (const float*)d_in[6];
    const float* b2   = (const float*)d_in[7];
    float* out        = (float*)d_out;

    const int N = in_sizes[0] / 128;
    const int E = in_sizes[1] / 2;
    const int* src_idx = ei;
    const int* dst_idx = ei + E;

    // Workspace layout: cnt [N] | agg [N*128] | h [N*128]
    char* ws   = (char*)d_ws;
    float* cnt = (float*)ws;
    float* agg = (float*)(ws + (((size_t)N * 4 + 255) & ~(size_t)255));
    float* h   = agg + (size_t)N * 128;

    const long aggN = (long)N * 128;
    const int  zeroBlocks  = (int)((aggN + 255) / 256);
    const int  cntBlocks   = (N + 255) / 256;
    const int  countBlocks = (E + 255) / 256;
    const int  scatBlocks  = (E * 32 + 255) / 256;     // one wave per edge
    const int  gemmWaves   = (N / 16) * 8;             // N divisible by 16 (50000)
    const int  gemmBlocks  = (gemmWaves * 32 + 255) / 256;

    // ---- Layer 1 ----
    sage_zero_kernel<<<cntBlocks, 256, 0, stream>>>(cnt, N);
    sage_zero_kernel<<<zeroBlocks, 256, 0, stream>>>(agg, aggN);
    sage_count_kernel<<<countBlocks, 256, 0, stream>>>(dst_idx, cnt, E);
    sage_scatter_kernel<<<scatBlocks, 256, 0, stream>>>(x, src_idx, dst_idx, agg, E);
    sage_gemm_wmma_kernel<<<gemmBlocks, 256, 0, stream>>>(agg, cnt, x, Wl1, Wr1, b1,
                                                          h, N, /*relu=*/1);
    // ---- Layer 2 ----
    sage_zero_kernel<<<zeroBlocks, 256, 0, stream>>>(agg, aggN);
    sage_scatter_kernel<<<scatBlocks, 256, 0, stream>>>(h, src_idx, dst_idx, agg, E);
    sage_gemm_wmma_kernel<<<gemmBlocks, 256, 0, stream>>>(agg, cnt, h, Wl2, Wr2, b2,
                                                          out, N, /*relu=*/0);
}